// TripCenterLoss_min_margin_47090021433771
// MI455X (gfx1250) — compile-verified
//
#include <hip/hip_runtime.h>
#include <hip/hip_bf16.h>
#include <stdint.h>

#define BATCH 2048
#define FEAT 256
#define NCLASS 512
#define ROWS_PER_BLOCK 8
#define TILE_C 64                    // centers per TDM tile
#define NTILES (NCLASS / TILE_C)     // 8
#define ROW_STRIDE (FEAT + 4)        // 260 floats: TDM pads 4 dwords after each 256-dword row

typedef unsigned int u32x4 __attribute__((ext_vector_type(4)));
typedef int i32x4 __attribute__((ext_vector_type(4)));
typedef int i32x8 __attribute__((ext_vector_type(8)));

// Issue a TDM load of a [TILE_C x FEAT] f32 tile (row-major, stride FEAT) from
// global `gsrc` into LDS at byte offset `lds_off`, with 4-dword row padding.
__device__ __forceinline__ void tdm_issue(const float* gsrc, unsigned lds_off) {
  uint64_t ga = (uint64_t)(uintptr_t)gsrc;
  u32x4 g0;
  g0.x = 1u;                                            // count=1 (valid), user mode
  g0.y = lds_off;                                       // lds_addr[31:0]
  g0.z = (unsigned)ga;                                  // global_addr[31:0]
  g0.w = (unsigned)((ga >> 32) & 0x01FFFFFFull)         // global_addr[56:32]
         | (2u << 30);                                  // type = 2 ("image")
  i32x8 g1;
  g1[0] = (int)((2u << 16)     // data_size = 2 -> 4 bytes
              | (1u << 20)     // pad_enable
              | (7u << 22)     // pad_interval = 7 -> every 256 dwords
              | (3u << 25));   // pad_amount = 3 -> 4 dwords
  g1[1] = (int)(((unsigned)FEAT & 0xFFFFu) << 16);                 // tensor_dim0[15:0]
  g1[2] = (int)((((unsigned)FEAT >> 16) & 0xFFFFu)                 // tensor_dim0[31:16]
              | (((unsigned)NCLASS & 0xFFFFu) << 16));             // tensor_dim1[15:0]
  g1[3] = (int)((((unsigned)NCLASS >> 16) & 0xFFFFu)               // tensor_dim1[31:16]
              | (((unsigned)FEAT & 0xFFFFu) << 16));               // tile_dim0 = 256
  g1[4] = (int)TILE_C;                                             // tile_dim1 = 64, tile_dim2 = 0
  g1[5] = (int)FEAT;                                               // tensor_dim0_stride[31:0]
  g1[6] = 0;                                                       // stride hi / dim1_stride lo
  g1[7] = 0;
  i32x4 gz = {0, 0, 0, 0};
#if __clang_major__ >= 23
  i32x8 gz8 = {0, 0, 0, 0, 0, 0, 0, 0};
  __builtin_amdgcn_tensor_load_to_lds(g0, g1, gz, gz, gz8, 0);
#else
  __builtin_amdgcn_tensor_load_to_lds(g0, g1, gz, gz, 0);
#endif
}

// Kernel 1: per-row own-class distance and min other-class distance.
// One wave (32 lanes) per batch row; 8 rows per block; centers streamed
// through LDS in double-buffered TDM tiles of 64 rows.
__global__ void __launch_bounds__(256)
dist_kernel(const float* __restrict__ x, const int* __restrict__ labels,
            const float* __restrict__ centers,
            float* __restrict__ dist, float* __restrict__ dmin) {
  __shared__ float tile[2][TILE_C * ROW_STRIDE];   // 2 x 66,560 B
  __shared__ float xs[ROWS_PER_BLOCK][FEAT];       // 8 KB

  const int tid  = (int)threadIdx.x;
  const int wave = tid >> 5;
  const int lane = tid & 31;
  const int row  = (int)blockIdx.x * ROWS_PER_BLOCK + wave;

  // Stage this wave's x row into LDS (2 x float4 per lane).
  {
    const float4* src = (const float4*)(x + (size_t)row * FEAT);
    float4* dst = (float4*)&xs[wave][0];
    dst[lane]      = src[lane];
    dst[lane + 32] = src[lane + 32];
  }

  const int lab = labels[row];

  if (wave == 0) {
    tdm_issue(centers, (unsigned)(uintptr_t)&tile[0][0]);  // prefetch tile 0
  }

  float own = 0.0f;
  float mn  = INFINITY;

  for (int t = 0; t < NTILES; ++t) {
    if (wave == 0) {
      if (t + 1 < NTILES) {
        tdm_issue(centers + (size_t)(t + 1) * TILE_C * FEAT,
                  (unsigned)(uintptr_t)&tile[(t + 1) & 1][0]);
        __builtin_amdgcn_s_wait_tensorcnt(1);  // tile t complete (in-order)
      } else {
        __builtin_amdgcn_s_wait_tensorcnt(0);  // last tile complete
      }
    }
    __syncthreads();  // tile t visible to all waves

    const float* cbase = &tile[t & 1][0];
    const float* c0 = cbase + (size_t)lane * ROW_STRIDE;
    const float* c1 = cbase + (size_t)(lane + 32) * ROW_STRIDE;
    float a0 = 0.0f, a1 = 0.0f;
#pragma unroll 4
    for (int d = 0; d < FEAT; d += 4) {
      float4 xv = *(const float4*)&xs[wave][d];    // wave-uniform broadcast
      float4 v0 = *(const float4*)&c0[d];
      float4 v1 = *(const float4*)&c1[d];
      a0 += fabsf(xv.x - v0.x) + fabsf(xv.y - v0.y) +
            fabsf(xv.z - v0.z) + fabsf(xv.w - v0.w);
      a1 += fabsf(xv.x - v1.x) + fabsf(xv.y - v1.y) +
            fabsf(xv.z - v1.z) + fabsf(xv.w - v1.w);
    }
    const int cg0 = t * TILE_C + lane;
    const int cg1 = cg0 + 32;
    if (cg0 == lab) own += a0; else mn = fminf(mn, a0);
    if (cg1 == lab) own += a1; else mn = fminf(mn, a1);
    __syncthreads();  // all waves done with tile t before TDM overwrites it
  }

  // wave32 reductions (sum for own: exactly one lane holds nonzero; min for mn)
  for (int m = 16; m >= 1; m >>= 1) {
    own += __shfl_xor(own, m, 32);
    mn = fminf(mn, __shfl_xor(mn, m, 32));
  }
  if (lane == 0) {
    dist[row] = own;
    dmin[row] = mn;
  }
}

// Kernel 2: global means -> dynamic margin -> loss.
__global__ void __launch_bounds__(256)
loss_kernel(const float* __restrict__ dist, const float* __restrict__ dmin,
            const float* __restrict__ margin, float* __restrict__ out) {
  __shared__ float s0[256];
  __shared__ float s1[256];
  __shared__ float dynm;
  const int tid = (int)threadIdx.x;

  float sd = 0.0f, sm = 0.0f;
  for (int i = tid; i < BATCH; i += 256) { sd += dist[i]; sm += dmin[i]; }
  s0[tid] = sd; s1[tid] = sm;
  __syncthreads();
  for (int off = 128; off >= 1; off >>= 1) {
    if (tid < off) { s0[tid] += s0[tid + off]; s1[tid] += s1[tid + off]; }
    __syncthreads();
  }
  if (tid == 0) {
    float delta = (s1[0] - s0[0]) / (float)BATCH;  // mean(dmin) - mean(dist)
    float m = margin[0];
    dynm = (delta > 0.0f) ? m * (1.0f + delta) : m;
  }
  __syncthreads();

  const float m = dynm;
  float acc = 0.0f;
  for (int i = tid; i < BATCH; i += 256) {
    acc += fmaxf(m + dist[i] - dmin[i], 0.0f);
  }
  s0[tid] = acc;
  __syncthreads();
  for (int off = 128; off >= 1; off >>= 1) {
    if (tid < off) s0[tid] += s0[tid + off];
    __syncthreads();
  }
  if (tid == 0) out[0] = s0[0] / (float)BATCH;
}

extern "C" void kernel_launch(void* const* d_in, const int* in_sizes, int n_in,
                              void* d_out, int out_size, void* d_ws, size_t ws_size,
                              hipStream_t stream) {
  (void)in_sizes; (void)n_in; (void)out_size; (void)ws_size;
  const float* x       = (const float*)d_in[0];
  const int*   labels  = (const int*)d_in[1];
  const float* margin  = (const float*)d_in[2];
  // d_in[3] = incenter_weight (unused by the reference loss value)
  const float* centers = (const float*)d_in[4];
  float* out = (float*)d_out;

  float* dist = (float*)d_ws;          // [BATCH]
  float* dmin = dist + BATCH;          // [BATCH]

  dist_kernel<<<BATCH / ROWS_PER_BLOCK, 256, 0, stream>>>(x, labels, centers, dist, dmin);
  loss_kernel<<<1, 256, 0, stream>>>(dist, dmin, margin, out);
}